// GCNModel_26018911879219
// MI455X (gfx1250) — compile-verified
//
#include <hip/hip_runtime.h>
#include <hip/hip_bf16.h>
#include <math.h>

// ---------------- problem constants (match reference) ----------------
#define N_NODES  100000
#define N_EDGES  1600000
#define N_GRAPHS 512
#define IN_FEATS 128
#define HIDDEN   128
#define EXTRA    8

typedef __attribute__((ext_vector_type(2))) float v2f;
typedef __attribute__((ext_vector_type(8))) float v8f;

__device__ __forceinline__ float selu_f(float x) {
    const float scale = 1.0507009873554805f;
    const float alpha = 1.6732632423543772f;
    return scale * (x > 0.0f ? x : alpha * (expf(x) - 1.0f));
}

// ---------------- degree accumulation: deg_out[src]+=1, deg_in[dst]+=1 ----
__global__ void deg_kernel(const int* __restrict__ esrc, const int* __restrict__ edst,
                           float* __restrict__ deg_src, float* __restrict__ deg_dst) {
    int e = blockIdx.x * blockDim.x + threadIdx.x;
    if (e >= N_EDGES) return;
    atomicAdd(deg_src + esrc[e], 1.0f);
    atomicAdd(deg_dst + edst[e], 1.0f);
}

// ---------------- deg -> clip(deg,1)^-0.5 in place ----------------
__global__ void norm_kernel(float* __restrict__ ns, float* __restrict__ nd) {
    int i = blockIdx.x * blockDim.x + threadIdx.x;
    if (i >= N_NODES) return;
    ns[i] = rsqrtf(fmaxf(ns[i], 1.0f));
    nd[i] = rsqrtf(fmaxf(nd[i], 1.0f));
}

// ---------------- edge scatter: agg[dst] += x[src] * norm_src[src] -------
// one wave (32 lanes) per edge; lane handles 4 contiguous floats (float4)
__global__ void scatter_edges(const float* __restrict__ x, const float* __restrict__ nsrc,
                              const int* __restrict__ esrc, const int* __restrict__ edst,
                              float* __restrict__ agg) {
    int gtid = blockIdx.x * blockDim.x + threadIdx.x;
    int e    = gtid >> 5;
    int lane = gtid & 31;
    if (e >= N_EDGES) return;
    int s = esrc[e];
    int d = edst[e];
    float ns = nsrc[s];
    const float4 v = *(const float4*)(x + (size_t)s * 128 + lane * 4);
    float* dp = agg + (size_t)d * 128 + lane * 4;
    atomicAdd(dp + 0, v.x * ns);
    atomicAdd(dp + 1, v.y * ns);
    atomicAdd(dp + 2, v.z * ns);
    atomicAdd(dp + 3, v.w * ns);
}

// ---------------- WMMA GEMM: out = act((rowScale .* A) @ W + b) ----------
// A: rows x 128 f32, W: 128 x 128 f32 row-major, out: rows x 128 f32.
// One wave computes a 16-row x 128-col slab with 8 independent 16x16 f32
// accumulators; K swept 4 at a time via V_WMMA_F32_16X16X4_F32 (exact f32).
// ISA A-layout (16x4 f32): lanes 0-15 hold {K=k,k+1}, lanes 16-31 {k+2,k+3}
// -> one aligned float2 load per lane per k-step, reused for all 8 N-tiles.
__global__ void gemm128_wmma(const float* __restrict__ A, const float* __restrict__ rowScale,
                             const float* __restrict__ W, const float* __restrict__ bias,
                             float* __restrict__ out, int nTiles, int doSelu) {
    int wavesPerBlk = blockDim.x >> 5;
    int tile = blockIdx.x * wavesPerBlk + (threadIdx.x >> 5);
    if (tile >= nTiles) return;                 // wave-uniform: EXEC stays all-1s
    int lane = threadIdx.x & 31;
    int half = lane >> 4;                       // 0: K=k,k+1   1: K=k+2,k+3
    int l16  = lane & 15;

    int arow = tile * 16 + l16;                 // A-fragment row for this lane
    float scl = rowScale ? rowScale[arow] : 1.0f;
    const float* Arow = A + (size_t)arow * 128;

    v8f acc[8] = {};                            // 8 N-tiles of 16 cols each

    for (int k = 0; k < 128; k += 4) {
        v2f a = *(const v2f*)(Arow + k + 2 * half);
        a *= scl;                               // fuse D_dst^{-1/2}
#pragma unroll
        for (int nt = 0; nt < 8; ++nt) {
            int col = nt * 16 + l16;
            v2f b;
            b.x = W[(size_t)(k + 2 * half)     * 128 + col];
            b.y = W[(size_t)(k + 2 * half + 1) * 128 + col];
            acc[nt] = __builtin_amdgcn_wmma_f32_16x16x4_f32(
                false, a, false, b, (short)0, acc[nt], false, false);
        }
    }

    // C/D layout: VGPR r -> row tile*16 + half*8 + r, col = nt*16 + l16
#pragma unroll
    for (int nt = 0; nt < 8; ++nt) {
        int col = nt * 16 + l16;
        float bv = bias[col];
#pragma unroll
        for (int r = 0; r < 8; ++r) {
            int orow = tile * 16 + half * 8 + r;
            float v = acc[nt][r] + bv;
            if (doSelu) v = selu_f(v);
            out[(size_t)orow * 128 + col] = v;
        }
    }
}

// ---------------- per-graph segment sum + count --------------------------
__global__ void readout_kernel(const float* __restrict__ x, const int* __restrict__ gids,
                               float* __restrict__ gsum, float* __restrict__ gcnt) {
    int gtid = blockIdx.x * blockDim.x + threadIdx.x;
    int node = gtid >> 5;
    int lane = gtid & 31;
    if (node >= N_NODES) return;
    int g = gids[node];
    const float4 v = *(const float4*)(x + (size_t)node * 128 + lane * 4);
    float* dp = gsum + (size_t)g * 128 + lane * 4;
    atomicAdd(dp + 0, v.x);
    atomicAdd(dp + 1, v.y);
    atomicAdd(dp + 2, v.z);
    atomicAdd(dp + 3, v.w);
    if (lane == 0) atomicAdd(gcnt + g, 1.0f);
}

__global__ void mean_kernel(const float* __restrict__ gsum, const float* __restrict__ gcnt,
                            float* __restrict__ emb) {
    int i = blockIdx.x * blockDim.x + threadIdx.x;
    if (i >= N_GRAPHS * 128) return;
    int g = i >> 7;
    emb[i] = gsum[i] / fmaxf(gcnt[g], 1.0f);
}

// ---------------- MLP head ----------------------------------------------
// y1 = selu([emb, feats_graph] @ Wl1 + bl1)      (512 x 256)
__global__ void head1_kernel(const float* __restrict__ emb, const float* __restrict__ fg,
                             const float* __restrict__ Wl1, const float* __restrict__ bl1,
                             float* __restrict__ y1) {
    int i = blockIdx.x * blockDim.x + threadIdx.x;
    if (i >= N_GRAPHS * 256) return;
    int g = i >> 8;
    int j = i & 255;
    float s = bl1[j];
    const float* eg = emb + (size_t)g * 128;
#pragma unroll 8
    for (int c = 0; c < 128; ++c) s += eg[c] * Wl1[(size_t)c * 256 + j];
    const float* fgg = fg + (size_t)g * EXTRA;
#pragma unroll
    for (int c = 0; c < EXTRA; ++c) s += fgg[c] * Wl1[(size_t)(128 + c) * 256 + j];
    y1[i] = selu_f(s);
}

// y2 = selu(y1 @ Wl2 + bl2)                      (512 x 128)
__global__ void head2_kernel(const float* __restrict__ y1, const float* __restrict__ Wl2,
                             const float* __restrict__ bl2, float* __restrict__ y2) {
    int i = blockIdx.x * blockDim.x + threadIdx.x;
    if (i >= N_GRAPHS * 128) return;
    int g = i >> 7;
    int j = i & 127;
    float s = bl2[j];
    const float* yg = y1 + (size_t)g * 256;
#pragma unroll 8
    for (int c = 0; c < 256; ++c) s += yg[c] * Wl2[(size_t)c * 128 + j];
    y2[i] = selu_f(s);
}

// out = y2 @ Wl3 + bl3                           (512)
__global__ void head3_kernel(const float* __restrict__ y2, const float* __restrict__ Wl3,
                             const float* __restrict__ bl3, float* __restrict__ out) {
    int g = blockIdx.x * blockDim.x + threadIdx.x;
    if (g >= N_GRAPHS) return;
    float s = bl3[0];
    const float* yg = y2 + (size_t)g * 128;
#pragma unroll 8
    for (int c = 0; c < 128; ++c) s += yg[c] * Wl3[c];
    out[g] = s;
}

// ---------------- launch --------------------------------------------------
extern "C" void kernel_launch(void* const* d_in, const int* in_sizes, int n_in,
                              void* d_out, int out_size, void* d_ws, size_t ws_size,
                              hipStream_t stream) {
    const float* feats_node  = (const float*)d_in[0];
    const float* feats_graph = (const float*)d_in[1];
    const int*   edge_src    = (const int*)d_in[2];
    const int*   edge_dst    = (const int*)d_in[3];
    const int*   graph_ids   = (const int*)d_in[4];
    const float* W1 = (const float*)d_in[5];
    const float* b1 = (const float*)d_in[6];
    const float* W2 = (const float*)d_in[7];
    const float* b2 = (const float*)d_in[8];
    const float* W3 = (const float*)d_in[9];
    const float* b3 = (const float*)d_in[10];
    const float* Wl1 = (const float*)d_in[11];
    const float* bl1 = (const float*)d_in[12];
    const float* Wl2 = (const float*)d_in[13];
    const float* bl2 = (const float*)d_in[14];
    const float* Wl3 = (const float*)d_in[15];
    const float* bl3 = (const float*)d_in[16];
    float* out = (float*)d_out;

    // workspace layout (floats)
    float* ws = (float*)d_ws;
    float* norm_src = ws;                              // 100000
    float* norm_dst = norm_src + N_NODES;              // 100000
    float* bufA     = norm_dst + N_NODES;              // 100000*128 (agg)
    float* bufB     = bufA + (size_t)N_NODES * 128;    // 100000*128 (x)
    float* gsum     = bufB + (size_t)N_NODES * 128;    // 512*128
    float* gcnt     = gsum + N_GRAPHS * 128;           // 512
    float* emb      = gcnt + N_GRAPHS;                 // 512*128
    float* y1       = emb  + N_GRAPHS * 128;           // 512*256
    float* y2       = y1   + N_GRAPHS * 256;           // 512*128

    const int nTiles = N_NODES / 16;                   // 6250, exact
    const dim3 gemmBlk(128);                           // 4 waves
    const dim3 gemmGrd((nTiles + 3) / 4);
    const int scatBlocks = (N_EDGES * 32 + 255) / 256;
    const size_t nodeBytes = (size_t)N_NODES * 128 * sizeof(float);

    // degrees -> norms
    hipMemsetAsync(norm_src, 0, 2 * N_NODES * sizeof(float), stream);
    deg_kernel<<<(N_EDGES + 255) / 256, 256, 0, stream>>>(edge_src, edge_dst, norm_src, norm_dst);
    norm_kernel<<<(N_NODES + 255) / 256, 256, 0, stream>>>(norm_src, norm_dst);

    // layer 1: feats_node -> bufB
    hipMemsetAsync(bufA, 0, nodeBytes, stream);
    scatter_edges<<<scatBlocks, 256, 0, stream>>>(feats_node, norm_src, edge_src, edge_dst, bufA);
    gemm128_wmma<<<gemmGrd, gemmBlk, 0, stream>>>(bufA, norm_dst, W1, b1, bufB, nTiles, 1);

    // layer 2: bufB -> bufB
    hipMemsetAsync(bufA, 0, nodeBytes, stream);
    scatter_edges<<<scatBlocks, 256, 0, stream>>>(bufB, norm_src, edge_src, edge_dst, bufA);
    gemm128_wmma<<<gemmGrd, gemmBlk, 0, stream>>>(bufA, norm_dst, W2, b2, bufB, nTiles, 1);

    // layer 3 (no selu): bufB -> bufB
    hipMemsetAsync(bufA, 0, nodeBytes, stream);
    scatter_edges<<<scatBlocks, 256, 0, stream>>>(bufB, norm_src, edge_src, edge_dst, bufA);
    gemm128_wmma<<<gemmGrd, gemmBlk, 0, stream>>>(bufA, norm_dst, W3, b3, bufB, nTiles, 0);

    // segment-mean readout
    hipMemsetAsync(gsum, 0, (N_GRAPHS * 128 + N_GRAPHS) * sizeof(float), stream);
    readout_kernel<<<(N_NODES * 32 + 255) / 256, 256, 0, stream>>>(bufB, graph_ids, gsum, gcnt);
    mean_kernel<<<(N_GRAPHS * 128 + 255) / 256, 256, 0, stream>>>(gsum, gcnt, emb);

    // MLP head
    head1_kernel<<<(N_GRAPHS * 256 + 255) / 256, 256, 0, stream>>>(emb, feats_graph, Wl1, bl1, y1);
    head2_kernel<<<(N_GRAPHS * 128 + 255) / 256, 256, 0, stream>>>(y1, Wl2, bl2, y2);
    head3_kernel<<<(N_GRAPHS + 255) / 256, 256, 0, stream>>>(y2, Wl3, bl3, out);
}